// gdGPT_37684043055675
// MI455X (gfx1250) — compile-verified
//
#include <hip/hip_runtime.h>

// Problem sizes
constexpr int kB = 8, kS = 1024, kD = 768, kH = 12, kV = 50257, kF = 3072;
constexpr int kNT = (kV + 15) / 16; // 3142 vocab tiles of 16

typedef float v8f __attribute__((ext_vector_type(8)));
typedef float v2f __attribute__((ext_vector_type(2)));

// workspace layout (floats)
constexpr int OFF_EMEAN = 0;
constexpr int OFF_ATTN  = OFF_EMEAN + kD;          // H*S softmax row per head
constexpr int OFF_BSUM  = OFF_ATTN + kH * kS;
constexpr int OFF_YHEAD = OFF_BSUM + kD;           // B*H*D
constexpr int OFF_YFIN  = OFF_YHEAD + kB * kH * kD;
constexpr int OFF_H1    = OFF_YFIN + kB * kD;
constexpr int OFF_X     = OFF_H1 + kB * kF;        // total ~124k floats (<0.5MB)

// ---- e_mean: deterministic column mean of wte (V x D) ----
__global__ void k_colmean(const float* __restrict__ wte, float* __restrict__ emean) {
    __shared__ float part[16][17];
    int c = threadIdx.x & 15, rg = threadIdx.x >> 4;
    int col = blockIdx.x * 16 + c;
    float s = 0.f;
    for (int r = rg; r < kV; r += 16) s += wte[(size_t)r * kD + col];
    part[rg][c] = s;
    __syncthreads();
    for (int k = 8; k > 0; k >>= 1) {
        if (rg < k) part[rg][c] += part[rg + k][c];
        __syncthreads();
    }
    if (rg == 0) emean[col] = part[0][c] * (1.0f / (float)kV);
}

// ---- attention softmax row at s = S-1 (batch independent) ----
__global__ void k_attn(const float* __restrict__ wpe, const float* __restrict__ wq,
                       const float* __restrict__ wk, float* __restrict__ attn) {
    __shared__ float qk[kD];
    __shared__ float red[256];
    int h = blockIdx.x, tid = threadIdx.x;
    for (int d = tid; d < kD; d += 256)
        qk[d] = wpe[(size_t)kS * kD + d] * wq[h * kD + d] * wk[h * kD + d];
    __syncthreads();
    float sc[4], lmax = -1e30f;
    for (int i = 0; i < 4; i++) {
        int t = tid + i * 256;
        const float* p = wpe + (size_t)t * kD;
        float s = 0.f;
        for (int d = 0; d < kD; d++) s += qk[d] * p[d];
        sc[i] = s;
        lmax = fmaxf(lmax, s);
    }
    red[tid] = lmax; __syncthreads();
    for (int k = 128; k > 0; k >>= 1) { if (tid < k) red[tid] = fmaxf(red[tid], red[tid + k]); __syncthreads(); }
    float m = red[0]; __syncthreads();
    float lsum = 0.f;
    for (int i = 0; i < 4; i++) { sc[i] = __expf(sc[i] - m); lsum += sc[i]; }
    red[tid] = lsum; __syncthreads();
    for (int k = 128; k > 0; k >>= 1) { if (tid < k) red[tid] += red[tid + k]; __syncthreads(); }
    float inv = 1.0f / red[0];
    for (int i = 0; i < 4; i++) attn[h * kS + tid + i * 256] = sc[i] * inv;
}

// ---- b_term sum over batch at s = S-1 ----
__global__ void k_bsum(const int* __restrict__ idx, const float* __restrict__ wte,
                       const float* __restrict__ emean, float* __restrict__ bsum) {
    int d = threadIdx.x;
    float s = 0.f;
    for (int b = 0; b < kB; b++)
        s += wte[(size_t)idx[b * kS + (kS - 1)] * kD + d];
    bsum[d] = (s - (float)kB * emean[d]) * (1.0f / (float)kS);
}

// ---- per-head value aggregation: yhead[b,h,d] ----
__global__ void __launch_bounds__(kD) k_yhead(const int* __restrict__ idx,
        const float* __restrict__ wte, const float* __restrict__ emean,
        const float* __restrict__ attn, const float* __restrict__ wv,
        const float* __restrict__ wlr, float* __restrict__ yhead) {
    __shared__ float a_s[kH * kS]; // 48KB
    int b = blockIdx.x, d = threadIdx.x;
    for (int i = d; i < kH * kS; i += kD) a_s[i] = attn[i];
    __syncthreads();
    float acc[kH];
#pragma unroll
    for (int h = 0; h < kH; h++) acc[h] = 0.f;
    const int* ib = idx + b * kS;
    for (int t = 0; t < kS; t++) {
        float ev = wte[(size_t)ib[t] * kD + d];
#pragma unroll
        for (int h = 0; h < kH; h++) acc[h] += a_s[h * kS + t] * ev;
    }
    float em = emean[d];
#pragma unroll
    for (int h = 0; h < kH; h++)
        yhead[(size_t)b * (kH * kD) + h * kD + d] =
            wv[h * kD + d] * wlr[h] * (1.0f / (float)kS) * (acc[h] - em);
}

// ---- output projection + b_term ----
__global__ void k_wo(const float* __restrict__ yhead, const float* __restrict__ wo,
                     const float* __restrict__ bsum, float* __restrict__ yfin) {
    int b = blockIdx.x;
    int dout = blockIdx.y * 256 + threadIdx.x;
    const float* yh = yhead + (size_t)b * (kH * kD);
    const float* wrow = wo + (size_t)dout * (kH * kD);
    float s = 0.f;
#pragma unroll 8
    for (int j = 0; j < kH * kD; j++) s += yh[j] * wrow[j];
    yfin[b * kD + dout] = s + bsum[dout];
}

// ---- MLP up + exact gelu ----
__global__ void k_mlp1(const float* __restrict__ yfin, const float* __restrict__ w1,
                       float* __restrict__ h1) {
    int b = blockIdx.x;
    int f = blockIdx.y * 256 + threadIdx.x;
    const float* y = yfin + b * kD;
    const float* wrow = w1 + (size_t)f * kD;
    float s = 0.f;
#pragma unroll 8
    for (int d = 0; d < kD; d++) s += y[d] * wrow[d];
    h1[b * kF + f] = 0.5f * s * (1.0f + erff(s * 0.70710678118654752f));
}

// ---- MLP down + residual ----
__global__ void k_mlp2(const float* __restrict__ yfin, const float* __restrict__ h1,
                       const float* __restrict__ w2, float* __restrict__ xout) {
    int b = blockIdx.x;
    int dout = blockIdx.y * 256 + threadIdx.x;
    const float* hb = h1 + b * kF;
    const float* wrow = w2 + (size_t)dout * kF;
    float s = 0.f;
#pragma unroll 8
    for (int f = 0; f < kF; f++) s += hb[f] * wrow[f];
    xout[b * kD + dout] = yfin[b * kD + dout] + s;
}

// ---- vocab projection: logits[b,v] = x[b,:] . wte[v,:], fp32 WMMA 16x16x4 ----
__global__ void __launch_bounds__(256) k_logits(const float* __restrict__ x,
        const float* __restrict__ wte, float* __restrict__ out) {
    __shared__ float xs[16 * kD]; // rows 0..7 = x, rows 8..15 zero (A-matrix padding)
    int tid = threadIdx.x;
    for (int i = tid; i < 16 * kD; i += 256)
        xs[i] = (i < kB * kD) ? x[i] : 0.f;
    __syncthreads();
    int wid = tid >> 5, lane = tid & 31;
    int tIdx = blockIdx.x * 8 + wid;
    if (tIdx >= kNT) return;           // wave-uniform: EXEC stays all-ones in live waves
    int n = lane & 15;                 // A: M index; B: N index
    int hh = lane >> 4;                // K-half selector
    int v0 = tIdx * 16;
    int vr = v0 + n; if (vr > kV - 1) vr = kV - 1;
    const float2* w2 = reinterpret_cast<const float2*>(wte);
    size_t browBase = (size_t)vr * (kD / 2);
    v8f c = {0.f, 0.f, 0.f, 0.f, 0.f, 0.f, 0.f, 0.f};
    for (int k = 0; k < kD; k += 4) {
        int kk = k + 2 * hh;
        v2f a, b;
        a.x = xs[n * kD + kk];
        a.y = xs[n * kD + kk + 1];
        float2 bv = w2[browBase + (kk >> 1)];
        b.x = bv.x; b.y = bv.y;
        c = __builtin_amdgcn_wmma_f32_16x16x4_f32(false, a, false, b,
                                                  (short)0, c, false, false);
    }
    if (hh == 0) {                     // lanes 0..15: C rows r = batch 0..7
        int v = v0 + n;
        if (v < kV) {
#pragma unroll
            for (int r = 0; r < kB; r++) out[(size_t)r * kV + v] = c[r];
        }
    }
}

extern "C" void kernel_launch(void* const* d_in, const int* in_sizes, int n_in,
                              void* d_out, int out_size, void* d_ws, size_t ws_size,
                              hipStream_t stream) {
    const int*   idx = (const int*)d_in[0];
    const float* wte = (const float*)d_in[1];
    const float* wpe = (const float*)d_in[2];
    const float* wq  = (const float*)d_in[3];
    const float* wk  = (const float*)d_in[4];
    const float* wv  = (const float*)d_in[5];
    const float* wlr = (const float*)d_in[6];
    const float* wo  = (const float*)d_in[7];
    const float* w1  = (const float*)d_in[8];
    const float* w2  = (const float*)d_in[9];
    float* out = (float*)d_out;
    float* ws  = (float*)d_ws;

    float* emean = ws + OFF_EMEAN;
    float* attn  = ws + OFF_ATTN;
    float* bsum  = ws + OFF_BSUM;
    float* yhead = ws + OFF_YHEAD;
    float* yfin  = ws + OFF_YFIN;
    float* h1    = ws + OFF_H1;
    float* x     = ws + OFF_X;

    k_colmean<<<dim3(kD / 16), dim3(256), 0, stream>>>(wte, emean);
    k_attn   <<<dim3(kH),      dim3(256), 0, stream>>>(wpe, wq, wk, attn);
    k_bsum   <<<dim3(1),       dim3(kD),  0, stream>>>(idx, wte, emean, bsum);
    k_yhead  <<<dim3(kB),      dim3(kD),  0, stream>>>(idx, wte, emean, attn, wv, wlr, yhead);
    k_wo     <<<dim3(kB, kD / 256),  dim3(256), 0, stream>>>(yhead, wo, bsum, yfin);
    k_mlp1   <<<dim3(kB, kF / 256),  dim3(256), 0, stream>>>(yfin, w1, h1);
    k_mlp2   <<<dim3(kB, kD / 256),  dim3(256), 0, stream>>>(yfin, h1, w2, x);
    k_logits <<<dim3((kNT + 7) / 8), dim3(256), 0, stream>>>(x, wte, out);
}